// OutBlock_62010737819928
// MI455X (gfx1250) — compile-verified
//
#include <hip/hip_runtime.h>
#include <hip/hip_bf16.h>

typedef __attribute__((ext_vector_type(2))) float v2f;
typedef __attribute__((ext_vector_type(8))) float v8f;

// Problem constants (from reference)
#define LATENT_DIM 512
#define IN_CH      64
#define NSPATIAL   (64 * 64 * 64)      // 262144 per (b, c)
#define NBATCH     8
#define TILES_PER_BATCH (NSPATIAL / 16) // 16384
#define TILES_TOTAL     (NBATCH * TILES_PER_BATCH)
// 0.01 / sqrt(512)
#define WEIGHT_GAIN 4.4194173824159224e-4f
#define LR_MULT 0.01f

// ---------------------------------------------------------------------------
// Kernel 1: mod[b][c] = (w[b,:]·fc_weight[c,:]) * gain + bias[c]*LR, times cw[c]
// Trivial: 8*64 outputs, 512-long dots (~0.5 MFLOP). One block, 512 threads.
// ---------------------------------------------------------------------------
__global__ __launch_bounds__(512) void compute_mod_kernel(
    const float* __restrict__ w, const float* __restrict__ fcw,
    const float* __restrict__ bias, const float* __restrict__ cw,
    float* __restrict__ mod)
{
    const int t = threadIdx.x;          // 0..511
    const int b = t >> 6;               // 0..7
    const int c = t & 63;               // 0..63
    const float* wr = w + b * LATENT_DIM;
    const float* fr = fcw + c * LATENT_DIM;
    float s = 0.0f;
    #pragma unroll 8
    for (int k = 0; k < LATENT_DIM; ++k)
        s = fmaf(wr[k], fr[k], s);
    const float style = s * WEIGHT_GAIN + bias[c] * LR_MULT;
    mod[t] = style * cw[c];
}

// ---------------------------------------------------------------------------
// Kernel 2: out[b, s] = sum_c mod[b,c] * x[b,c,s]  via V_WMMA_F32_16X16X4_F32
//
// Per WMMA step k (channels c0 = 4k .. 4k+3):
//   B (4x16): lane L: half = L>>4, n = L&15
//     vgpr0 = x[b][c0 + 2*half    ][s0 + n]     (K = 0 | 2)
//     vgpr1 = x[b][c0 + 2*half + 1][s0 + n]     (K = 1 | 3)
//   A (16x4): mod broadcast over all 16 M rows:
//     vgpr0 = mod[b][c0 + 2*half], vgpr1 = mod[b][c0 + 2*half + 1]
// => every row of D equals the channel dot for spatial tile [s0, s0+16).
// Result read from acc[0], lanes 0..15 (M=0 row).
// ---------------------------------------------------------------------------
__global__ __launch_bounds__(256) void mod_reduce_wmma_kernel(
    const float* __restrict__ x, const float* __restrict__ mod,
    float* __restrict__ out)
{
    const int lane   = threadIdx.x & 31;
    const int waveId = blockIdx.x * (blockDim.x >> 5) + (threadIdx.x >> 5);
    const int half   = lane >> 4;      // 0: lanes 0-15, 1: lanes 16-31
    const int nidx   = lane & 15;

    // 8192 waves, 16 contiguous tiles each; 1024 waves per batch (exact).
    const int b = waveId >> 10;
    const float* xb = x + (size_t)b * IN_CH * (size_t)NSPATIAL;

    // Build the 16 A-operand steps once per wave (b is fixed for the wave).
    const float m_lo = mod[b * 64 + lane];        // channels  0..31
    const float m_hi = mod[b * 64 + 32 + lane];   // channels 32..63
    float a0v[16], a1v[16];
    #pragma unroll
    for (int k = 0; k < 16; ++k) {
        const int c0 = 4 * k;
        const float src = (c0 < 32) ? m_lo : m_hi;   // uniform select
        const int base = (c0 & 31) + 2 * half;
        a0v[k] = __shfl(src, base, 32);
        a1v[k] = __shfl(src, base + 1, 32);
    }

    const int tile0 = waveId * 16;
    #pragma unroll 1
    for (int i = 0; i < 16; ++i) {
        const int tile = tile0 + i;
        const int s0 = (tile & (TILES_PER_BATCH - 1)) << 4;
        // lane's base pointer: channel (2*half), spatial s0+nidx
        const float* px = xb + (size_t)(2 * half) * NSPATIAL + s0 + nidx;

        v8f acc = {};
        #pragma unroll
        for (int k = 0; k < 16; ++k) {
            const float* p = px + (size_t)(4 * k) * NSPATIAL;
            v2f bm, am;
            bm.x = __builtin_nontemporal_load(p);            // channel c0+2h
            bm.y = __builtin_nontemporal_load(p + NSPATIAL); // channel c0+2h+1
            am.x = a0v[k];
            am.y = a1v[k];
            acc = __builtin_amdgcn_wmma_f32_16x16x4_f32(
                /*neg_a=*/false, am, /*neg_b=*/false, bm,
                /*c_mod=*/(short)0, acc, /*reuse_a=*/false, /*reuse_b=*/false);
        }

        // D row M=0: acc[0], lanes 0..15 hold out[s0 + n]
        if (lane < 16) {
            out[(size_t)b * NSPATIAL + s0 + lane] = acc[0];
        }
    }
}

// ---------------------------------------------------------------------------
// Launch: inputs in dict order: x, w, fc_weight, fc_bias, conv_weight
// ---------------------------------------------------------------------------
extern "C" void kernel_launch(void* const* d_in, const int* in_sizes, int n_in,
                              void* d_out, int out_size, void* d_ws, size_t ws_size,
                              hipStream_t stream) {
    const float* x    = (const float*)d_in[0];
    const float* w    = (const float*)d_in[1];
    const float* fcw  = (const float*)d_in[2];
    const float* bias = (const float*)d_in[3];
    const float* cw   = (const float*)d_in[4];
    float* out = (float*)d_out;
    float* mod = (float*)d_ws;   // 512 floats (2 KB) of scratch

    compute_mod_kernel<<<1, 512, 0, stream>>>(w, fcw, bias, cw, mod);

    // 1024 blocks * 8 waves = 8192 waves; 16 tiles each = 131072 tiles total.
    mod_reduce_wmma_kernel<<<1024, 256, 0, stream>>>(x, mod, out);
}